// SDETGNNModel_13554916786267
// MI455X (gfx1250) — compile-verified
//
#include <hip/hip_runtime.h>
#include <math.h>

#define NPTS   8192
#define DIN    83
#define H      256
#define C_OUT  34
#define TD     16
#define KNN    11
#define NSTEPS 10
#define NLAYERS 2
#define KC     64          // K-chunk staged in LDS per GEMM block iteration

typedef __attribute__((ext_vector_type(2))) float v2f;
typedef __attribute__((ext_vector_type(8))) float v8f;

// ---------------------------------------------------------------------------
// Fused GEMM:  C = epilogue( A[M,K] @ W[K,Nc] + bias + b2s*bias2 )
//   act: 0=none 1=relu 2=tanh 3=sigmoid ; if resid: C = resid + act_out*rs
// Block = 128 threads (4 waves) computes a 32-row x 64-col output tile.
// Each wave owns one 16-col strip and two 16x16 accumulators (two row
// groups), reusing its B fragment. A and W are staged in LDS per 64-deep
// K-chunk with zero-fill for K/Nc edges, so the inner WMMA loop has NO
// branches and NO global loads. LDS strides padded to 68 floats so the
// 16-lane row-strided A reads hit 16 distinct banks.
// ---------------------------------------------------------------------------
__global__ void __launch_bounds__(128)
gemm_wmma_f32(const float* __restrict__ A, const float* __restrict__ W,
              const float* __restrict__ bias,
              const float* __restrict__ bias2, float bias2_scale,
              const float* resid, float res_scale,
              float* Cout, int Nc, int K, int act)
{
    __shared__ float lA[32 * 68];
    __shared__ float lB[KC * 68];
    const int tid  = threadIdx.x;
    const int row0 = blockIdx.y * 32;
    const int col_base = blockIdx.x * 64;
    const int lda = 68, ldb = 68;

    const int wave  = tid >> 5;
    const int lane  = tid & 31;
    const int mrow  = lane & 15;
    const int khalf = (lane < 16) ? 0 : 2;
    const int colL  = wave * 16 + (lane & 15);   // local col 0..63
    const int bcol  = col_base + colL;
    const bool colOK = bcol < Nc;

    v8f acc0 = {0.f, 0.f, 0.f, 0.f, 0.f, 0.f, 0.f, 0.f};
    v8f acc1 = {0.f, 0.f, 0.f, 0.f, 0.f, 0.f, 0.f, 0.f};

    for (int k0 = 0; k0 < K; k0 += KC) {
        // ---- stage A tile: 32 rows x KC (zero-filled past K) ----
        for (int i = tid; i < 32 * KC; i += 128) {
            int r = i >> 6, c = i & 63;
            int kg = k0 + c;
            lA[r * lda + c] = (kg < K) ? A[(size_t)(row0 + r) * K + kg] : 0.0f;
        }
        // ---- stage W tile: KC x 64 (zero-filled past K / Nc) ----
        for (int i = tid; i < KC * 64; i += 128) {
            int r = i >> 6, c = i & 63;
            int kg = k0 + r;
            int cg = col_base + c;
            lB[r * ldb + c] = (kg < K && cg < Nc) ? W[(size_t)kg * Nc + cg] : 0.0f;
        }
        __syncthreads();

#pragma unroll
        for (int kk = 0; kk < KC; kk += 4) {
            v2f a0, a1, b;
            a0.x = lA[mrow * lda + kk + khalf];
            a0.y = lA[mrow * lda + kk + khalf + 1];
            a1.x = lA[(mrow + 16) * lda + kk + khalf];
            a1.y = lA[(mrow + 16) * lda + kk + khalf + 1];
            b.x  = lB[(kk + khalf) * ldb + colL];
            b.y  = lB[(kk + khalf + 1) * ldb + colL];
            acc0 = __builtin_amdgcn_wmma_f32_16x16x4_f32(false, a0, false, b,
                                                         (short)0, acc0, false, false);
            acc1 = __builtin_amdgcn_wmma_f32_16x16x4_f32(false, a1, false, b,
                                                         (short)0, acc1, false, false);
        }
        __syncthreads();
    }

    if (colOK) {
        float bval = bias ? bias[bcol] : 0.0f;
        if (bias2) bval += bias2_scale * bias2[bcol];
        v8f accs[2] = {acc0, acc1};
#pragma unroll
        for (int g = 0; g < 2; ++g) {
#pragma unroll
            for (int v = 0; v < 8; ++v) {
                int m = g * 16 + v + ((lane >> 4) << 3);
                size_t off = (size_t)(row0 + m) * Nc + bcol;
                float val = accs[g][v] + bval;
                if (act == 1)      val = fmaxf(val, 0.0f);
                else if (act == 2) val = tanhf(val);
                else if (act == 3) val = 1.0f / (1.0f + __expf(-val));
                if (resid) val = resid[off] + val * res_scale;
                Cout[off] = val;
            }
        }
    }
}

// ---------------------------------------------------------------------------
// Fused similarity + top-k:  sim = hn @ hn^T  (never materialized).
// Block = 64 threads (2 waves) owns 16 rows; loops over column chunks of 32.
// Each wave computes one 16x16 WMMA tile per chunk into LDS; threads 0..15
// keep a running sorted top-11 (value,index) per row, then emit the
// row-normalized adjacency values + indices.
// ---------------------------------------------------------------------------
__global__ void __launch_bounds__(64)
simtopk_wmma(const float* __restrict__ hn,
             float* __restrict__ aval, int* __restrict__ aidx)
{
    __shared__ float lA[16 * 260];
    __shared__ float lB[32 * 260];
    __shared__ float lS[16 * 32];
    const int tid  = threadIdx.x;
    const int row0 = blockIdx.x * 16;
    const int lda  = 260;

    for (int i = tid; i < 16 * H; i += 64) {
        int r = i >> 8, c = i & 255;
        lA[r * lda + c] = hn[(size_t)(row0 + r) * H + c];
    }

    float tv[KNN]; int ti[KNN];
#pragma unroll
    for (int j = 0; j < KNN; ++j) { tv[j] = -2.0f; ti[j] = 0; }

    const int wave  = tid >> 5;
    const int lane  = tid & 31;
    const int mrow  = lane & 15;
    const int khalf = (lane < 16) ? 0 : 2;
    const int cb    = wave * 16 + (lane & 15);   // local column 0..31

    for (int c0 = 0; c0 < NPTS; c0 += 32) {
        for (int i = tid; i < 32 * H; i += 64) {
            int r = i >> 8, c = i & 255;
            lB[r * lda + c] = hn[(size_t)(c0 + r) * H + c];
        }
        __syncthreads();

        v8f acc = {0.f, 0.f, 0.f, 0.f, 0.f, 0.f, 0.f, 0.f};
#pragma unroll 8
        for (int kk = 0; kk < H; kk += 4) {
            v2f a, b;
            a.x = lA[mrow * lda + kk + khalf];
            a.y = lA[mrow * lda + kk + khalf + 1];
            b.x = lB[cb * lda + kk + khalf];
            b.y = lB[cb * lda + kk + khalf + 1];
            acc = __builtin_amdgcn_wmma_f32_16x16x4_f32(false, a, false, b,
                                                        (short)0, acc, false, false);
        }
#pragma unroll
        for (int v = 0; v < 8; ++v) {
            int m = v + ((lane >> 4) << 3);
            lS[m * 32 + cb] = acc[v];
        }
        __syncthreads();

        if (tid < 16) {
            for (int j = 0; j < 32; ++j) {
                float val = lS[tid * 32 + j];
                if (val > tv[KNN - 1]) {
                    int p = KNN - 1;
                    while (p > 0 && val > tv[p - 1]) {
                        tv[p] = tv[p - 1]; ti[p] = ti[p - 1]; --p;
                    }
                    tv[p] = val; ti[p] = c0 + j;
                }
            }
        }
        __syncthreads();   // protects lS / orders next chunk's compute
    }

    if (tid < 16) {
        float s = 0.0f;
        for (int j = 0; j < KNN; ++j) s += tv[j];
        float denom = fmaxf(s, 1.0f);
        for (int j = 0; j < KNN; ++j) {
            aval[(size_t)(row0 + tid) * KNN + j] = tv[j] / denom;
            aidx[(size_t)(row0 + tid) * KNN + j] = ti[j];
        }
    }
}

// ---------------------------------------------------------------------------
// Small elementwise / reduction kernels
// ---------------------------------------------------------------------------
__global__ void bn_kernel(const float* __restrict__ in, float* __restrict__ out,
                          const float* g, const float* b,
                          const float* m, const float* v)
{
    int i = blockIdx.x * blockDim.x + threadIdx.x;
    int c = i & (H - 1);
    out[i] = (in[i] - m[c]) * rsqrtf(v[c] + 1e-5f) * g[c] + b[c];
}

__global__ void __launch_bounds__(256)
rownorm_kernel(const float* __restrict__ h, float* __restrict__ hn)
{
    __shared__ float red[256];
    int row = blockIdx.x, t = threadIdx.x;
    float x = h[(size_t)row * H + t];
    red[t] = x * x;
    __syncthreads();
    for (int s = 128; s > 0; s >>= 1) {
        if (t < s) red[t] += red[t + s];
        __syncthreads();
    }
    float denom = fmaxf(sqrtf(red[0]), 1e-12f);
    hn[(size_t)row * H + t] = x / denom;
}

__global__ void __launch_bounds__(256)
agg_kernel(const float* __restrict__ aval, const int* __restrict__ aidx,
           const float* __restrict__ ht, float* __restrict__ agg)
{
    __shared__ float sv[KNN];
    __shared__ int   si[KNN];
    int row = blockIdx.x, t = threadIdx.x;
    if (t < KNN) { sv[t] = aval[row * KNN + t]; si[t] = aidx[row * KNN + t]; }
    __syncthreads();
    float s = 0.0f;
#pragma unroll
    for (int m = 0; m < KNN; ++m) s += sv[m] * ht[(size_t)si[m] * H + t];
    agg[(size_t)row * H + t] = s;
}

// te = MLP(t) [1,H]; gvec[j] = gate_b[j] + sum_k te[k]*gate_w[H+k][j]
__global__ void __launch_bounds__(256)
tegate_kernel(float t, const float* __restrict__ w1, const float* __restrict__ b1,
              const float* __restrict__ w2, const float* __restrict__ b2,
              const float* __restrict__ gate_w, const float* __restrict__ gate_b,
              float* __restrict__ gvec)
{
    __shared__ float hid[TD];
    __shared__ float te[H];
    int tx = threadIdx.x;
    if (tx < TD) hid[tx] = fmaxf(t * w1[tx] + b1[tx], 0.0f);
    __syncthreads();
    float s = b2[tx];
#pragma unroll
    for (int k = 0; k < TD; ++k) s += hid[k] * w2[k * H + tx];
    te[tx] = s;
    __syncthreads();
    float gv = gate_b[tx];
    for (int k = 0; k < H; ++k) gv += te[k] * gate_w[(size_t)(H + k) * H + tx];
    gvec[tx] = gv;
}

__global__ void combine_kernel(const float* __restrict__ g,
                               const float* __restrict__ agg,
                               const float* __restrict__ ht,
                               float* __restrict__ hout)
{
    size_t i = (size_t)blockIdx.x * blockDim.x + threadIdx.x;
    float gg = g[i];
    hout[i] = fmaxf(gg * agg[i] + (1.0f - gg) * ht[i], 0.0f);
}

// ---------------------------------------------------------------------------
extern "C" void kernel_launch(void* const* d_in, const int* in_sizes, int n_in,
                              void* d_out, int out_size, void* d_ws, size_t ws_size,
                              hipStream_t stream)
{
    (void)in_sizes; (void)n_in; (void)out_size; (void)ws_size;

    const float* x         = (const float*)d_in[0];
    const float* enc_w1    = (const float*)d_in[1];
    const float* enc_b1    = (const float*)d_in[2];
    const float* enc_w2    = (const float*)d_in[3];
    const float* enc_b2    = (const float*)d_in[4];
    const float* bn_g      = (const float*)d_in[5];
    const float* bn_b      = (const float*)d_in[6];
    const float* bn_m      = (const float*)d_in[7];
    const float* bn_v      = (const float*)d_in[8];
    const float* tg_node_w = (const float*)d_in[9];
    const float* tg_node_b = (const float*)d_in[10];
    const float* tg_t_w1   = (const float*)d_in[11];
    const float* tg_t_b1   = (const float*)d_in[12];
    const float* tg_t_w2   = (const float*)d_in[13];
    const float* tg_t_b2   = (const float*)d_in[14];
    const float* tg_gate_w = (const float*)d_in[15];
    const float* tg_gate_b = (const float*)d_in[16];
    const float* sde_w1    = (const float*)d_in[17];
    const float* sde_b1    = (const float*)d_in[18];
    const float* sde_w2    = (const float*)d_in[19];
    const float* sde_b2    = (const float*)d_in[20];
    const float* sde_w3    = (const float*)d_in[21];
    const float* sde_b3    = (const float*)d_in[22];
    const float* dec_w1    = (const float*)d_in[23];
    const float* dec_b1    = (const float*)d_in[24];
    const float* dec_w2    = (const float*)d_in[25];
    const float* dec_b2    = (const float*)d_in[26];

    float* ws   = (float*)d_ws;
    float* hA   = ws;                            // [N,H]
    float* hB   = hA + (size_t)NPTS * H;         // [N,H]
    float* c3   = hB + (size_t)NPTS * H;         // [N,H]
    float* c4   = c3 + (size_t)NPTS * H;         // [N,H]
    float* aval = c4 + (size_t)NPTS * H;         // [N,KNN]
    int*   aidx = (int*)(aval + (size_t)NPTS * KNN);
    float* gvec = (float*)(aidx + (size_t)NPTS * KNN);

    dim3 blk(128);
    auto gemm = [&](const float* A, const float* W, const float* bias,
                    const float* bias2, float b2s, const float* resid, float rs,
                    float* Cout, int Nc, int K, int act) {
        dim3 grid((Nc + 63) / 64, NPTS / 32);
        gemm_wmma_f32<<<grid, blk, 0, stream>>>(A, W, bias, bias2, b2s,
                                                resid, rs, Cout, Nc, K, act);
    };

    // --- encoder ---
    gemm(x,  enc_w1, enc_b1, nullptr, 0.f, nullptr, 0.f, hA, H, DIN, 1);
    gemm(hA, enc_w2, enc_b2, nullptr, 0.f, nullptr, 0.f, hB, H, H,   1);
    bn_kernel<<<(NPTS * H) / 256, 256, 0, stream>>>(hB, hA, bn_g, bn_b, bn_m, bn_v);

    // --- kNN graph (fused sim + top-k) ---
    rownorm_kernel<<<NPTS, 256, 0, stream>>>(hA, hB);
    simtopk_wmma<<<NPTS / 16, 64, 0, stream>>>(hB, aval, aidx);

    // --- temporal gated graph conv layers ---
    float t = 0.0f;
    for (int l = 0; l < NLAYERS; ++l) {
        gemm(hA, tg_node_w + (size_t)l * H * H, tg_node_b + l * H,
             nullptr, 0.f, nullptr, 0.f, hB, H, H, 0);                 // ht
        agg_kernel<<<NPTS, 256, 0, stream>>>(aval, aidx, hB, c3);      // agg
        tegate_kernel<<<1, 256, 0, stream>>>(t,
            tg_t_w1 + l * TD, tg_t_b1 + l * TD,
            tg_t_w2 + (size_t)l * TD * H, tg_t_b2 + l * H,
            tg_gate_w + (size_t)l * 2 * H * H, tg_gate_b + l * H, gvec);
        gemm(c3, tg_gate_w + (size_t)l * 2 * H * H, nullptr,
             gvec, 1.0f, nullptr, 0.f, c4, H, H, 3);                   // gate
        combine_kernel<<<(NPTS * H) / 256, 256, 0, stream>>>(c4, c3, hB, hA);
        t += 0.5f;
    }

    // --- SDE drift-only Euler steps ---
    const float dt = 1.0f / NSTEPS;
    for (int s = 0; s < NSTEPS; ++s) {
        float ts = s * dt;
        gemm(hA, sde_w1, sde_b1, sde_w1 + (size_t)H * H, ts,
             nullptr, 0.f, hB, H, H, 2);                               // tanh
        gemm(hB, sde_w2, sde_b2, nullptr, 0.f, nullptr, 0.f, c3, H, H, 2);
        gemm(c3, sde_w3, sde_b3, nullptr, 0.f, hA, dt, hA, H, H, 2);   // h += tanh*dt
    }

    // --- decoder ---
    gemm(hA, dec_w1, dec_b1, nullptr, 0.f, nullptr, 0.f, hB, H / 2, H, 1);
    gemm(hB, dec_w2, dec_b2, nullptr, 0.f, nullptr, 0.f,
         (float*)d_out, C_OUT, H / 2, 0);
}